// StandardAngular_35768487641382
// MI455X (gfx1250) — compile-verified
//
#include <hip/hip_runtime.h>
#include <math.h>

// ---------------------------------------------------------------------------
// ANI angular symmetry-function term, MI455X (gfx1250).
//
// Roofline: 2M pairs * (32B in + 128B out) = 320MB @ 23.3TB/s ~= 14us.
// Memory-bound, store-heavy -> NT b128 stores (output is 256MB write-once,
// must not thrash the 192MB L2 which can hold the whole 64MB input).
// No contraction dimension -> WMMA not applicable; CDNA5 paths used instead:
// async global->LDS (ASYNCcnt) constant staging, s_wait_asynccnt,
// global_prefetch_b8.
// ---------------------------------------------------------------------------

#define AS1g __attribute__((address_space(1)))
#define AS3l __attribute__((address_space(3)))

typedef float v4f __attribute__((ext_vector_type(4)));

#define PI_F 3.14159265358979323846f
#define CUTOFF_F 3.5f

__global__ __launch_bounds__(256, 8)
void ani_angular_kernel(const float* __restrict__ vec,       // (2, P, 3)
                        const float* __restrict__ dist,      // (2, P)
                        const float* __restrict__ eta_p,     // (1,)
                        const float* __restrict__ zeta_p,    // (1,)
                        const float* __restrict__ shifts_p,  // (8,)
                        const float* __restrict__ sect_p,    // (4,)
                        float* __restrict__ out,             // (P, 32)
                        int P)
{
    __shared__ float s_const[12];   // [0..7] shifts, [8..11] angle sections
    const int tid = threadIdx.x;

    // ---- Stage the 12 small constants into LDS via the gfx1250 async
    //      global->LDS path (tracked by ASYNCcnt), with a plain-load fallback.
#if __has_builtin(__builtin_amdgcn_global_load_async_to_lds_b32)
    if (tid < 12) {
        const float* src = (tid < 8) ? (shifts_p + tid) : (sect_p + (tid - 8));
        __builtin_amdgcn_global_load_async_to_lds_b32(
            (AS1g int*)src, (AS3l int*)&s_const[tid], 0, 0);
    }
#if __has_builtin(__builtin_amdgcn_s_wait_asynccnt)
    __builtin_amdgcn_s_wait_asynccnt(0);
#else
    asm volatile("s_wait_asynccnt 0" ::: "memory");
#endif
#else
    if (tid < 12) {
        s_const[tid] = (tid < 8) ? shifts_p[tid] : sect_p[tid - 8];
    }
#endif
    __syncthreads();

    // Uniform scalars (compiler turns these into s_loads).
    const float eta    = eta_p[0];
    const float zeta   = zeta_p[0];
    const float negEta = -eta;
    const float kfc    = PI_F / CUTOFF_F;

    // Per-thread copies of constants; sin/cos of the 4 sections computed once
    // per thread and amortized over the grid-stride loop (~4 pairs/thread).
    float shf[8], cosTh[4], sinTh[4];
#pragma unroll
    for (int i = 0; i < 8; ++i) shf[i] = s_const[i];
#pragma unroll
    for (int a = 0; a < 4; ++a) {
        const float th = s_const[8 + a];
        cosTh[a] = __cosf(th);
        sinTh[a] = __sinf(th);
    }

    const float* __restrict__ v0  = vec;                    // (P,3)
    const float* __restrict__ v1  = vec + 3 * (size_t)P;    // (P,3)
    const float* __restrict__ d0p = dist;                   // (P,)
    const float* __restrict__ d1p = dist + (size_t)P;       // (P,)

    const int stride = gridDim.x * blockDim.x;
    for (int p = blockIdx.x * blockDim.x + tid; p < P; p += stride) {
        // Prefetch next grid-stride tile (gfx1250 global_prefetch_b8).
        const int pn = p + stride;
        if (pn < P) {
            __builtin_prefetch(v0 + 3 * (size_t)pn, 0, 3);
            __builtin_prefetch(v1 + 3 * (size_t)pn, 0, 3);
            __builtin_prefetch(d0p + pn, 0, 3);
            __builtin_prefetch(d1p + pn, 0, 3);
        }

        const size_t p3 = 3 * (size_t)p;
        const float ax = v0[p3 + 0], ay = v0[p3 + 1], az = v0[p3 + 2];
        const float bx = v1[p3 + 0], by = v1[p3 + 1], bz = v1[p3 + 2];
        const float d0 = d0p[p];
        const float d1 = d1p[p];

        // cos(angle) after the 0.95 shrink; sin via identity (arccos-free).
        const float dot   = __fmaf_rn(ax, bx, __fmaf_rn(ay, by, az * bz));
        const float denom = fmaxf(d0 * d1, 1e-10f);
        const float c     = 0.95f * __fdividef(dot, denom);
        const float s     = sqrtf(1.0f - c * c);   // NaN if |c|>1, as reference

        // cutoff factor (incl. the global 2.0)
        const float fc0  = __fmaf_rn(0.5f, __cosf(d0 * kfc), 0.5f);
        const float fc1  = __fmaf_rn(0.5f, __cosf(d1 * kfc), 0.5f);
        const float pref = 2.0f * fc0 * fc1;

        const float m = 0.5f * (d0 + d1);

        // factor1[a] = ((1 + cos(angle - theta_a))/2)^zeta
        //            = exp(zeta * log(0.5 + 0.5*(c*cosT + s*sinT)))
        float f1[4];
#pragma unroll
        for (int a = 0; a < 4; ++a) {
            const float cd   = __fmaf_rn(c, cosTh[a], s * sinTh[a]);
            float base       = __fmaf_rn(0.5f, cd, 0.5f);
            base             = (base < 0.0f) ? 0.0f : base;  // keep NaN as NaN
            f1[a]            = __expf(zeta * __logf(base));
        }

        // factor2[s] * pref, fused into four-wide NT stores: out[p, s*4 + a]
        v4f* __restrict__ o = (v4f*)(out + 32 * (size_t)p);
#pragma unroll
        for (int si = 0; si < 8; ++si) {
            const float t  = m - shf[si];
            const float f2 = pref * __expf(negEta * t * t);
            v4f r;
            r.x = f2 * f1[0];
            r.y = f2 * f1[1];
            r.z = f2 * f1[2];
            r.w = f2 * f1[3];
            __builtin_nontemporal_store(r, o + si);  // stream: don't pollute L2
        }
    }
}

extern "C" void kernel_launch(void* const* d_in, const int* in_sizes, int n_in,
                              void* d_out, int out_size, void* d_ws, size_t ws_size,
                              hipStream_t stream) {
    const float* vec      = (const float*)d_in[0];  // vectors12 (2,P,3)
    const float* dist     = (const float*)d_in[1];  // distances12 (2,P)
    const float* eta      = (const float*)d_in[2];  // (1,)
    const float* zeta     = (const float*)d_in[3];  // (1,)
    const float* shifts   = (const float*)d_in[4];  // (8,)
    const float* sections = (const float*)d_in[5];  // (4,)
    float* out = (float*)d_out;

    const int P = in_sizes[0] / 6;   // 2 * P * 3 floats in vectors12

    const int threads = 256;                 // 8 wave32 per block
    const int ppt     = 4;                   // pairs per thread (grid-stride)
    int blocks = (P + threads * ppt - 1) / (threads * ppt);
    if (blocks < 1) blocks = 1;

    hipLaunchKernelGGL(ani_angular_kernel, dim3(blocks), dim3(threads), 0, stream,
                       vec, dist, eta, zeta, shifts, sections, out, P);
}